// EdgeAwareAttention_11441792876964
// MI455X (gfx1250) — compile-verified
//
#include <hip/hip_runtime.h>
#include <hip/hip_bf16.h>

#define N_NODES  50000
#define N_EDGES  400000
#define HIDDEN   128
#define N_HEADS  8
#define HEAD_DIM 16
#define EDGE_FEAT 16

typedef __attribute__((ext_vector_type(2))) float v2f;
typedef __attribute__((ext_vector_type(8))) float v8f;

// ---------------------------------------------------------------------------
// Exact fp32 WMMA GEMM: C[M=16*grid, 128] = A[M,128] * W[128,128]
// One wave computes one 16x16 tile with V_WMMA_F32_16X16X4_F32, K-loop of 32.
// Block = 256 threads = 8 waves; wave w owns output columns [16w, 16w+16).
// Layouts per CDNA5 ISA 7.12.2:
//   A 16x4 : lanes 0-15 hold M=0..15, khalf=0 ; lanes 16-31 khalf=2 ;
//            VGPR0 = K=khalf, VGPR1 = K=khalf+1
//   C/D    : VGPR r -> (M = r + 8*(lane>=16), N = lane&15)
// ---------------------------------------------------------------------------
__global__ __launch_bounds__(256) void gemm128_wmma_f32(
    const float* __restrict__ A,
    const float* __restrict__ W,
    float* __restrict__ C)
{
    const int lane  = threadIdx.x & 31;
    const int wave  = threadIdx.x >> 5;          // 0..7  -> n-tile
    const int m0    = blockIdx.x << 4;           // row tile (3125 tiles, exact)
    const int n0    = wave << 4;
    const int mrow  = m0 + (lane & 15);
    const int ncol  = n0 + (lane & 15);
    const int khalf = (lane >> 4) << 1;          // 0 or 2

    v8f acc = {};
#pragma unroll 4
    for (int k = 0; k < HIDDEN; k += 4) {
        v2f a, b;
        a.x = A[mrow * HIDDEN + k + khalf];
        a.y = A[mrow * HIDDEN + k + khalf + 1];
        b.x = W[(k + khalf)     * HIDDEN + ncol];
        b.y = W[(k + khalf + 1) * HIDDEN + ncol];
        acc = __builtin_amdgcn_wmma_f32_16x16x4_f32(
                  /*neg_a=*/false, a, /*neg_b=*/false, b,
                  /*c_mod=*/(short)0, acc, /*reuse_a=*/false, /*reuse_b=*/false);
    }

    const int mbase = m0 + ((lane >> 4) << 3);   // +0 or +8
#pragma unroll
    for (int r = 0; r < 8; ++r)
        C[(mbase + r) * HIDDEN + ncol] = acc[r];
}

// ---------------------------------------------------------------------------
// Monotone uint encoding of float so unsigned atomicMax == float max
// ---------------------------------------------------------------------------
__device__ __forceinline__ unsigned fkey(float f) {
    unsigned b = __float_as_uint(f);
    return (b & 0x80000000u) ? ~b : (b | 0x80000000u);
}
__device__ __forceinline__ float funkey(unsigned k) {
    unsigned b = (k & 0x80000000u) ? (k & 0x7fffffffu) : ~k;
    return __uint_as_float(b);
}

// ---------------------------------------------------------------------------
// scores[e,h] = (Q[src[e],h,:] . K[dst[e],h,:]) / sqrt(16) + ef[e] . Wb[:,h] + bb[h]
// fused atomic segment-max into mkey[src[e],h]
// ---------------------------------------------------------------------------
__global__ void scores_kernel(
    const float* __restrict__ Q, const float* __restrict__ K,
    const int* __restrict__ src, const int* __restrict__ dst,
    const float* __restrict__ ef, const float* __restrict__ Wb,
    const float* __restrict__ bb,
    float* __restrict__ scores, unsigned* __restrict__ mkey)
{
    int idx = blockIdx.x * blockDim.x + threadIdx.x;
    if (idx >= N_EDGES * N_HEADS) return;
    const int e = idx >> 3, h = idx & 7;
    const int s = src[e], d = dst[e];

    const float4* q = (const float4*)(Q + (size_t)s * HIDDEN + h * HEAD_DIM);
    const float4* k = (const float4*)(K + (size_t)d * HIDDEN + h * HEAD_DIM);
    float acc = 0.f;
#pragma unroll
    for (int i = 0; i < 4; ++i) {
        float4 a = q[i], b = k[i];
        acc += a.x * b.x + a.y * b.y + a.z * b.z + a.w * b.w;
    }
    acc *= 0.25f;                       // 1/sqrt(HEAD_DIM)

    float bias = bb[h];
    const float* f = ef + (size_t)e * EDGE_FEAT;
#pragma unroll
    for (int i = 0; i < EDGE_FEAT; ++i)
        bias += f[i] * Wb[i * N_HEADS + h];

    const float sc = acc + bias;
    scores[idx] = sc;
    atomicMax(&mkey[s * N_HEADS + h], fkey(sc));
}

// ---------------------------------------------------------------------------
// ex = exp(score - m[src]); scores <- ex; denom[src,h] += ex
// ---------------------------------------------------------------------------
__global__ void ex_kernel(
    const int* __restrict__ src,
    float* __restrict__ scores,
    const unsigned* __restrict__ mkey,
    float* __restrict__ denom)
{
    int idx = blockIdx.x * blockDim.x + threadIdx.x;
    if (idx >= N_EDGES * N_HEADS) return;
    const int e = idx >> 3, h = idx & 7;
    const int s = src[e];
    const float m  = funkey(mkey[s * N_HEADS + h]);
    const float ex = __expf(scores[idx] - m);
    scores[idx] = ex;
    atomicAdd(&denom[s * N_HEADS + h], ex);
}

// ---------------------------------------------------------------------------
// One 128-thread block per edge. Recompute vmod = ef@Wvm + bvm on the fly,
// attn = ex/denom[src], scatter attn*(V[dst]+vmod) into agg[src] with f32
// atomics (agg is 25.6 MB -> L2 resident).
// ---------------------------------------------------------------------------
__global__ __launch_bounds__(128) void agg_kernel(
    const int* __restrict__ src, const int* __restrict__ dst,
    const float* __restrict__ V,
    const float* __restrict__ ef, const float* __restrict__ Wvm,
    const float* __restrict__ bvm,
    const float* __restrict__ exbuf, const float* __restrict__ denom,
    float* __restrict__ agg)
{
    const int e   = blockIdx.x;
    const int dim = threadIdx.x;          // 0..127
    const int h   = dim >> 4;

    __shared__ float efs[EDGE_FEAT];
    if (dim < EDGE_FEAT) efs[dim] = ef[(size_t)e * EDGE_FEAT + dim];
    __syncthreads();

    const int s = src[e], d = dst[e];

    float vm = bvm[dim];
#pragma unroll
    for (int k = 0; k < EDGE_FEAT; ++k)
        vm += efs[k] * Wvm[k * HIDDEN + dim];

    const float ve = V[(size_t)d * HIDDEN + dim] + vm;
    const float a  = exbuf[(size_t)e * N_HEADS + h] /
                     denom[(size_t)s * N_HEADS + h];
    atomicAdd(&agg[(size_t)s * HIDDEN + dim], a * ve);
}

// ---------------------------------------------------------------------------
// Host side
// ---------------------------------------------------------------------------
extern "C" void kernel_launch(void* const* d_in, const int* in_sizes, int n_in,
                              void* d_out, int out_size, void* d_ws, size_t ws_size,
                              hipStream_t stream)
{
    const float* X   = (const float*)d_in[0];
    const int*   ei  = (const int*)  d_in[1];
    const float* EF  = (const float*)d_in[2];
    const float* Wq  = (const float*)d_in[3];
    const float* Wk  = (const float*)d_in[4];
    const float* Wv  = (const float*)d_in[5];
    const float* Wo  = (const float*)d_in[6];
    const float* Wb  = (const float*)d_in[7];
    const float* bb  = (const float*)d_in[8];
    const float* Wvm = (const float*)d_in[9];
    const float* bvm = (const float*)d_in[10];
    float* out = (float*)d_out;

    const int* src = ei;
    const int* dst = ei + N_EDGES;

    // workspace layout (floats)
    const size_t NODE = (size_t)N_NODES * HIDDEN;      // 6,400,000
    float* Q      = (float*)d_ws;
    float* Kb     = Q  + NODE;
    float* Vb     = Kb + NODE;
    float* AGG    = Vb + NODE;
    float* SCORES = AGG + NODE;                        // N_EDGES*N_HEADS
    float* DENOM  = SCORES + (size_t)N_EDGES * N_HEADS;
    unsigned* MKEY = (unsigned*)(DENOM + (size_t)N_NODES * N_HEADS);

    const int M_TILES = N_NODES / 16;                  // 3125, exact

    // 1) Q/K/V projections (exact fp32 WMMA)
    gemm128_wmma_f32<<<M_TILES, 256, 0, stream>>>(X, Wq, Q);
    gemm128_wmma_f32<<<M_TILES, 256, 0, stream>>>(X, Wk, Kb);
    gemm128_wmma_f32<<<M_TILES, 256, 0, stream>>>(X, Wv, Vb);

    // 2) zero AGG + SCORES + DENOM + MKEY (contiguous); uint 0 == -inf key
    const size_t zero_bytes =
        (NODE + (size_t)N_EDGES * N_HEADS + 2 * (size_t)N_NODES * N_HEADS) * 4;
    hipMemsetAsync(AGG, 0, zero_bytes, stream);

    // 3) scores + fused segment-max
    const int EH = N_EDGES * N_HEADS;
    scores_kernel<<<(EH + 255) / 256, 256, 0, stream>>>(
        Q, Kb, src, dst, EF, Wb, bb, SCORES, MKEY);

    // 4) exp + segment-sum of denominators
    ex_kernel<<<(EH + 255) / 256, 256, 0, stream>>>(src, SCORES, MKEY, DENOM);

    // 5) weighted scatter-aggregate
    agg_kernel<<<N_EDGES, 128, 0, stream>>>(
        src, dst, Vb, EF, Wvm, bvm, SCORES, DENOM, AGG);

    // 6) output projection
    gemm128_wmma_f32<<<M_TILES, 256, 0, stream>>>(AGG, Wo, out);
}